// PchipF0UpsamplerTorch_18219251270368
// MI455X (gfx1250) — compile-verified
//
#include <hip/hip_runtime.h>
#include <cstdint>

#define BB 16
#define TT 2048
#define LL (TT * 128)

// ---------------------------------------------------------------------------
// Phase 1: one block per batch. Stage f0 row via CDNA5 async global->LDS,
// compact voiced samples (stable), build PCHIP slopes in f64, emit:
//   X[b][T] (f64 knot x), Y[b][T] (f64 knot y), D[b][T] (f64 slopes),
//   LUT[b][T] (segment index for each integer floor(up_x)),
//   MASK[b][64] (voiced bitmask), NV[b] (#voiced).
// ---------------------------------------------------------------------------
__global__ __launch_bounds__(256) void pchip_prep(
    const float* __restrict__ f0,
    double* __restrict__ X, double* __restrict__ Y, double* __restrict__ D,
    int* __restrict__ LUT, unsigned* __restrict__ MASK, int* __restrict__ NV)
{
    __shared__ double   s_x[TT];
    __shared__ double   s_y[TT];
    __shared__ double   s_d[TT];      // first 8KB aliased as f32 f0 staging
    __shared__ int      s_xi[TT];
    __shared__ int      s_scan[256];
    __shared__ unsigned s_mask[64];

    const int t = threadIdx.x;
    const int b = blockIdx.x;
    float* s_f0 = (float*)s_d;

    // ---- async-load f0 row into LDS (ASYNCcnt path, GVS addressing) ----
    {
        unsigned long long base = (unsigned long long)(f0 + (size_t)b * TT);
        #pragma unroll
        for (int it = 0; it < 8; ++it) {
            int i = it * 256 + t;
            unsigned ldsoff = (unsigned)(uintptr_t)(s_f0 + i);
            unsigned voff   = (unsigned)(i * 4);
            asm volatile("global_load_async_to_lds_b32 %0, %1, %2"
                         :: "v"(ldsoff), "v"(voff), "s"(base) : "memory");
        }
        asm volatile("s_wait_asynccnt 0" ::: "memory");
    }
    if (t < 64) s_mask[t] = 0u;
    __syncthreads();

    // ---- voiced flags + bitmask + per-thread count (8 contiguous elems) ----
    float fv[8];
    unsigned bits = 0u;
    #pragma unroll
    for (int k = 0; k < 8; ++k) {
        fv[k] = s_f0[t * 8 + k];
        if (fv[k] > 0.0f) bits |= (1u << k);
    }
    int c = __popc(bits);
    atomicOr(&s_mask[(t * 8) >> 5], bits << ((t * 8) & 31));
    s_scan[t] = c;
    __syncthreads();

    // ---- inclusive scan over 256 thread counts ----
    for (int off = 1; off < 256; off <<= 1) {
        int v   = s_scan[t];
        int add = (t >= off) ? s_scan[t - off] : 0;
        __syncthreads();
        s_scan[t] = v + add;
        __syncthreads();
    }
    const int nv = s_scan[255];
    int run = s_scan[t] - c;   // exclusive prefix (stable compaction offset)

    // ---- stable compaction of voiced positions/values ----
    #pragma unroll
    for (int k = 0; k < 8; ++k) {
        if (bits & (1u << k)) {
            int i = t * 8 + k;
            s_xi[run] = i;
            s_x[run]  = (double)i;
            s_y[run]  = (double)fv[k];
            run++;
        }
    }
    __syncthreads();

    // ---- extension tail: x_last + (i+1-nv), y_last ----
    int xlast_i; double ylast;
    if (nv > 0) { xlast_i = s_xi[nv - 1]; ylast = s_y[nv - 1]; }
    else        { xlast_i = 0;            ylast = (double)s_f0[0]; }
    for (int i = nv + t; i < TT; i += 256) {
        int xv  = xlast_i + (i + 1 - nv);
        s_xi[i] = xv;
        s_x[i]  = (double)xv;
        s_y[i]  = ylast;
    }
    __syncthreads();

    // ---- interior PCHIP slopes (overwrites f0 staging; no longer needed) ----
    #pragma unroll
    for (int k = 0; k < 8; ++k) {
        int i = t * 8 + k;
        double dd = 0.0;
        if (i >= 1 && i <= TT - 2) {
            double xm = s_x[i - 1], x0 = s_x[i], xp = s_x[i + 1];
            double ym = s_y[i - 1], y0 = s_y[i], yp = s_y[i + 1];
            double hkm1 = x0 - xm, hk = xp - x0;
            double dkm1 = (y0 - ym) / hkm1;
            double dk   = (yp - y0) / hk;
            if (dkm1 * dk > 0.0) {
                double w1 = 2.0 * hk + hkm1, w2 = hk + 2.0 * hkm1;
                dd = (w1 + w2) / (w1 / dkm1 + w2 / dk);
            }
        }
        s_d[i] = dd;
    }
    __syncthreads();

    // ---- endpoint slopes (one-sided three-point, monotonicity clamped) ----
    if (t == 0) {
        double h0 = s_x[1] - s_x[0], h1 = s_x[2] - s_x[1];
        double del0 = (s_y[1] - s_y[0]) / h0;
        double del1 = (s_y[2] - s_y[1]) / h1;
        double d0 = ((2.0 * h0 + h1) * del0 - h0 * del1) / (h0 + h1);
        if (d0 * del0 <= 0.0) d0 = 0.0;
        else if ((del0 * del1 < 0.0) && (fabs(d0) > 3.0 * fabs(del0))) d0 = 3.0 * del0;

        int im1 = min(max(nv - 2, 0), TT - 2);
        int im2 = min(max(nv - 3, 0), TT - 2);
        double hm1 = s_x[im1 + 1] - s_x[im1], hm2 = s_x[im2 + 1] - s_x[im2];
        double dm1 = (s_y[im1 + 1] - s_y[im1]) / hm1;
        double dm2 = (s_y[im2 + 1] - s_y[im2]) / hm2;
        double dn = ((2.0 * hm1 + hm2) * dm1 - hm1 * dm2) / (hm1 + hm2);
        if (dn * dm1 <= 0.0) dn = 0.0;
        else if ((dm1 * dm2 < 0.0) && (fabs(dn) > 3.0 * fabs(dm1))) dn = 3.0 * dm1;

        if (nv == 2) { d0 = del0; dn = del0; }
        s_d[0] = d0;                                  // matches ref write order
        int jn = min(max(nv - 1, 0), TT - 1);
        s_d[jn] = dn;                                 // dn overwrites last
        NV[b] = nv;
    }
    if (t < 64) MASK[b * 64 + t] = s_mask[t];
    __syncthreads();

    // ---- LUT (seg for each integer floor) + table writeback ----
    const int segmax = max(nv - 2, 0);
    for (int i = t; i < TT; i += 256) {
        // upper_bound over strictly increasing s_xi
        int lo = 0, hi = TT;
        while (lo < hi) { int mid = (lo + hi) >> 1; if (s_xi[mid] <= i) lo = mid + 1; else hi = mid; }
        int seg = lo - 1;
        seg = min(max(seg, 0), segmax);
        LUT[b * TT + i] = seg;
        X[b * TT + i] = s_x[i];
        Y[b * TT + i] = s_y[i];
        D[b * TT + i] = s_d[i];
    }
}

// ---------------------------------------------------------------------------
// Phase 2: streaming Hermite evaluation, 4 outputs per thread, float4 stores.
// Bandwidth-bound on the 16 MB output write; tables hit L2 (~1 MB total).
// ---------------------------------------------------------------------------
__global__ __launch_bounds__(256) void pchip_eval(
    float* __restrict__ out,
    const double* __restrict__ X, const double* __restrict__ Y,
    const double* __restrict__ D, const int* __restrict__ LUT,
    const unsigned* __restrict__ MASK, const int* __restrict__ NV)
{
    const double STEP = (double)(TT - 1) / (double)(LL - 1);
    int gq = blockIdx.x * 256 + threadIdx.x;   // quad index
    int b  = gq >> 16;                         // 65536 quads per batch
    int q  = gq & 0xFFFF;
    int j0 = q * 4;

    const int nv = NV[b];
    const bool nvok = (nv >= 2);
    const double*  Xb = X + b * TT;
    const double*  Yb = Y + b * TT;
    const double*  Db = D + b * TT;
    const int*     Lb = LUT + b * TT;
    const unsigned* Mb = MASK + b * 64;

    float rr[4];
    #pragma unroll
    for (int k = 0; k < 4; ++k) {
        int j = j0 + k;
        double ux = (double)j * STEP;
        float v = 0.0f;
        if (nvok) {
            int back = __float2int_rn((float)ux);    // round half to even
            back = min(max(back, 0), TT - 1);
            if ((Mb[back >> 5] >> (back & 31)) & 1u) {
                int m = (int)ux;                     // floor (ux >= 0)
                if (m > TT - 1) m = TT - 1;
                int seg = Lb[m];
                double x0 = Xb[seg], x1 = Xb[seg + 1];
                double y0 = Yb[seg], y1 = Yb[seg + 1];
                double g0 = Db[seg], g1 = Db[seg + 1];
                double h  = x1 - x0;
                double tt = (ux - x0) / h;
                double t2 = tt * tt, t3 = t2 * tt;
                double up = (2.0 * t3 - 3.0 * t2 + 1.0) * y0
                          + (t3 - 2.0 * t2 + tt) * h * g0
                          + (-2.0 * t3 + 3.0 * t2) * y1
                          + (t3 - t2) * h * g1;
                v = fmaxf((float)up, 0.0f);
            }
        }
        rr[k] = v;
    }
    *(float4*)(out + (size_t)b * LL + j0) = make_float4(rr[0], rr[1], rr[2], rr[3]);
}

// ---------------------------------------------------------------------------
extern "C" void kernel_launch(void* const* d_in, const int* in_sizes, int n_in,
                              void* d_out, int out_size, void* d_ws, size_t ws_size,
                              hipStream_t stream) {
    const float* f0 = (const float*)d_in[0];
    float* out = (float*)d_out;

    char* ws = (char*)d_ws;
    double* X = (double*)ws;
    double* Y = X + BB * TT;
    double* D = Y + BB * TT;
    int* LUT = (int*)(D + BB * TT);
    unsigned* MASK = (unsigned*)(LUT + BB * TT);
    int* NV = (int*)(MASK + BB * 64);

    pchip_prep<<<BB, 256, 0, stream>>>(f0, X, Y, D, LUT, MASK, NV);
    pchip_eval<<<(BB * (LL / 4)) / 256, 256, 0, stream>>>(out, X, Y, D, LUT, MASK, NV);
}